// GConvLSTMModel_42150809043553
// MI455X (gfx1250) — compile-verified
//
#include <hip/hip_runtime.h>
#include <hip/hip_bf16.h>
#include <cstdint>

typedef float v2f __attribute__((ext_vector_type(2)));
typedef float v8f __attribute__((ext_vector_type(8)));

// Fixed slot stride (floats) between packed gate weight matrices; byte offset
// 204800 / 409600 fits the signed 24-bit VMEM instruction offset.
#define GATE_STRIDE 51200  // 320 * 160

// ---------------------------------------------------------------- utilities

__global__ void zero_int_kernel(int* __restrict__ p, unsigned n) {
  unsigned i = blockIdx.x * blockDim.x + threadIdx.x;
  if (i < n) p[i] = 0;
}

// degi[dst] += 1 for every edge (segment_sum over dst)
__global__ void degree_kernel(const int* __restrict__ dst, int* __restrict__ degi, int E) {
  int e = blockIdx.x * blockDim.x + threadIdx.x;
  if (e < E) atomicAdd(&degi[dst[e]], 1);
}

// w[e] = dinv[src]*dinv[dst], dinv = deg>0 ? rsqrt(deg) : 0  (deg>=1 when >0)
__global__ void edge_weight_kernel(const int* __restrict__ src, const int* __restrict__ dst,
                                   const int* __restrict__ degi, float* __restrict__ w, int E) {
  int e = blockIdx.x * blockDim.x + threadIdx.x;
  if (e < E) {
    int ds = degi[src[e]];
    int dd = degi[dst[e]];
    float is = ds > 0 ? rsqrtf((float)ds) : 0.0f;
    float id = dd > 0 ? rsqrtf((float)dd) : 0.0f;
    w[e] = is * id;
  }
}

// Single-workgroup exclusive scan of degi -> rowptr (1024-thread Hillis-Steele
// per 1024-chunk with running carry; ~100 chunks for N=100k).
__global__ __launch_bounds__(1024)
void scan_rowptr_kernel(const int* __restrict__ degi, int* __restrict__ rowptr, int N) {
  __shared__ int sdata[1024];
  __shared__ int carry_s;
  const int tid = threadIdx.x;
  if (tid == 0) carry_s = 0;
  __syncthreads();
  for (int base = 0; base < N; base += 1024) {
    int i = base + tid;
    int v = (i < N) ? degi[i] : 0;
    sdata[tid] = v;
    __syncthreads();
    for (int s = 1; s < 1024; s <<= 1) {
      int t = (tid >= s) ? sdata[tid - s] : 0;
      __syncthreads();
      sdata[tid] += t;
      __syncthreads();
    }
    int incl = sdata[tid];
    int carry = carry_s;
    if (i < N) rowptr[i] = carry + incl - v;  // exclusive
    __syncthreads();
    if (tid == 1023) carry_s = carry + sdata[1023];
    __syncthreads();
  }
  if (tid == 0) rowptr[N] = carry_s;
}

// Fill CSR: csr[slot] = (bitcast src, w). Order within a row is unordered
// (fp segment-sum order is unspecified in the reference as well).
__global__ void csr_fill_kernel(const int* __restrict__ src, const int* __restrict__ dst,
                                const float* __restrict__ ew, const int* __restrict__ rowptr,
                                int* __restrict__ cursor, v2f* __restrict__ csr, int E) {
  int e = blockIdx.x * blockDim.x + threadIdx.x;
  if (e < E) {
    int n = dst[e];
    int slot = rowptr[n] + atomicAdd(&cursor[n], 1);
    v2f sw;
    sw.x = __int_as_float(src[e]);
    sw.y = ew[e];
    csr[slot] = sw;
  }
}

// TxCat[n, 0:F] = h[n, 0:F]   (slot 0 of the concatenated Cheb buffer)
__global__ void copy_slot0_kernel(const float* __restrict__ h, float* __restrict__ TxCat,
                                  unsigned total, unsigned F, unsigned RD) {
  unsigned t = blockIdx.x * blockDim.x + threadIdx.x;
  if (t < total) {
    unsigned n = t / F;
    unsigned c = t - n * F;
    TxCat[(size_t)n * RD + c] = h[t];
  }
}

// Gather SpMM fused with Cheb combine:
//   Tx[n, koff_dst + c] = alpha * sum_{e in row n} Tx[src_e, koff_src + c] * w_e
//                         - (has_sub ? Tx[n, koff_sub + c] : 0)
// (alpha = -1 for k==1, -2 for k>=2: Lhat = -A_norm). Channel is the fast index:
// the F lanes sharing a row broadcast csr[e] and read the Tx row contiguously.
__global__ void spmm_gather_kernel(const v2f* __restrict__ csr, const int* __restrict__ rowptr,
                                   float* __restrict__ TxCat, unsigned RD,
                                   unsigned koff_src, unsigned koff_dst, unsigned koff_sub,
                                   int has_sub, float alpha, unsigned F, unsigned total) {
  unsigned t = blockIdx.x * blockDim.x + threadIdx.x;
  if (t >= total) return;
  unsigned n = t / F;
  unsigned c = t - n * F;
  int e0 = rowptr[n];
  int e1 = rowptr[n + 1];
  float sum = 0.0f;
  for (int e = e0; e < e1; ++e) {
    v2f sw = csr[e];
    unsigned s = (unsigned)__float_as_int(sw.x);
    sum = fmaf(TxCat[(size_t)s * RD + koff_src + c], sw.y, sum);
  }
  float v = alpha * sum;
  if (has_sub) v -= TxCat[(size_t)n * RD + koff_sub + c];
  TxCat[(size_t)n * RD + koff_dst + c] = v;
}

// Pack W [RD, Fout] -> Wp pair-interleaved + column-padded:
//   Wp[(r>>1)*Fp*2 + c*2 + (r&1)] = (c < Fout) ? W[r*Fout + c] : 0
// so a WMMA B fragment (rows k0+2*half+{0,1}, col c) is one aligned b64 load.
__global__ void pack_weights_kernel(const float* __restrict__ W, float* __restrict__ Wp,
                                    unsigned RD, unsigned Fout, unsigned Fp) {
  unsigned t = blockIdx.x * blockDim.x + threadIdx.x;
  unsigned total = RD * Fp;
  if (t < total) {
    unsigned r = t / Fp;
    unsigned c = t - r * Fp;
    float v = (c < Fout) ? W[r * Fout + c] : 0.0f;
    Wp[((r >> 1) * Fp + c) * 2 + (r & 1)] = v;
  }
}

// ------------------------------------------------- fused 3-gate WMMA GEMM
// 128 threads = 4 waves per block; each wave computes one 16x16 tile
// (nodes x out-channels) for gates i, c, o with V_WMMA_F32_16X16X4_F32, then
// applies the LSTM nonlinearity + ReLU. f-gate skipped (multiplies C==0).
// Hot loop: 1 A b64 load (immediate offsets), 1 B pointer bump, 3 B b64 loads
// at compile-time gate offsets, 3 WMMAs. No predication anywhere before the
// guarded stores, so EXEC stays all-ones as WMMA requires.

__device__ __forceinline__ float sigmoidf_(float x) { return 1.0f / (1.0f + __expf(-x)); }

__global__ __launch_bounds__(128)
void gate_gemm_wmma(const float* __restrict__ TxCat, int RD,
                    const float* __restrict__ Wp,   // [3][GATE_STRIDE] packed i,c,o
                    const float* __restrict__ bxi, const float* __restrict__ bhi,
                    const float* __restrict__ bbi,
                    const float* __restrict__ bxc, const float* __restrict__ bhc,
                    const float* __restrict__ bbc,
                    const float* __restrict__ bxo, const float* __restrict__ bho,
                    const float* __restrict__ bbo,
                    float* __restrict__ hout, int N, int Fout, int Fp) {
  const int wave = threadIdx.x >> 5;
  const int lane = threadIdx.x & 31;
  const int half = lane >> 4;   // 0: lanes 0-15, 1: lanes 16-31
  const int l16  = lane & 15;
  const int n0 = (blockIdx.x * 4 + wave) * 16;
  const int f0 = blockIdx.y * 16;
  const int fcol = f0 + l16;    // always < Fp (gridDim.y == Fp/16)

  // A fragment: row n0+l16 (clamped), K values k0 + 2*half + {0,1} -> one b64
  int arow_n = n0 + l16;
  if (arow_n >= N) arow_n = N - 1;
  const float* aptr = TxCat + (size_t)arow_n * RD + (half << 1);

  // Single B cursor; gates i/c/o at fixed GATE_STRIDE immediate offsets.
  const float* bp = Wp + ((size_t)((unsigned)half * (unsigned)Fp + (unsigned)fcol) * 2u);
  const size_t bstep = (size_t)Fp * 4u;  // 2 interleaved pairs per k-step

  v8f acc_i = {};
  v8f acc_c = {};
  v8f acc_o = {};

  for (int k0 = 0; k0 < RD; k0 += 4) {
    v2f a  = *(const v2f*)(aptr + k0);
    v2f bi = *(const v2f*)(bp);
    v2f bc = *(const v2f*)(bp + GATE_STRIDE);
    v2f bo = *(const v2f*)(bp + 2 * GATE_STRIDE);
    bp += bstep;

    // 8 args: (neg_a, A, neg_b, B, c_mod, C, reuse_a, reuse_b)
    acc_i = __builtin_amdgcn_wmma_f32_16x16x4_f32(false, a, false, bi, (short)0, acc_i, false, false);
    acc_c = __builtin_amdgcn_wmma_f32_16x16x4_f32(false, a, false, bc, (short)0, acc_c, false, false);
    acc_o = __builtin_amdgcn_wmma_f32_16x16x4_f32(false, a, false, bo, (short)0, acc_o, false, false);
  }

  const bool colok = (fcol < Fout);
  const int  fb = colok ? fcol : 0;  // clamped bias index (unused when !colok)
  const float bsi = bxi[fb] + bhi[fb] + bbi[fb];
  const float bsc = bxc[fb] + bhc[fb] + bbc[fb];
  const float bso = bxo[fb] + bho[fb] + bbo[fb];

#pragma unroll
  for (int v = 0; v < 8; ++v) {
    int node = n0 + v + (half << 3);  // C/D layout: m = v + 8*half
    if (colok && node < N) {
      float iv = sigmoidf_(acc_i[v] + bsi);
      float cv = iv * tanhf(acc_c[v] + bsc);
      float hv = sigmoidf_(acc_o[v] + bso) * tanhf(cv);
      hout[(size_t)node * Fout + fcol] = fmaxf(hv, 0.0f);
    }
  }
}

// ------------------------------------------------- final linear + softmax (2 classes)
__global__ void linear_softmax_kernel(const float* __restrict__ h, const float* __restrict__ W,
                                      const float* __restrict__ b, float* __restrict__ out,
                                      int N, int F) {
  int n = blockIdx.x * blockDim.x + threadIdx.x;
  if (n >= N) return;
  const float* hr = h + (size_t)n * F;
  float z0 = b[0], z1 = b[1];
  for (int c = 0; c < F; ++c) {
    float hv = hr[c];
    z0 = fmaf(hv, W[2 * c + 0], z0);
    z1 = fmaf(hv, W[2 * c + 1], z1);
  }
  float m = fmaxf(z0, z1);
  float e0 = __expf(z0 - m);
  float e1 = __expf(z1 - m);
  float inv = 1.0f / (e0 + e1);
  out[2 * n + 0] = e0 * inv;
  out[2 * n + 1] = e1 * inv;
}

// ---------------------------------------------------------------- launch

extern "C" void kernel_launch(void* const* d_in, const int* in_sizes, int n_in,
                              void* d_out, int out_size, void* d_ws, size_t ws_size,
                              hipStream_t stream) {
  const float* x    = (const float*)d_in[0];
  const int*   eidx = (const int*)d_in[1];
  const int N = in_sizes[0] / 10;
  const int E = in_sizes[1] / 2;
  const int* esrc = eidx;
  const int* edst = eidx + E;

  // workspace carve (256B aligned)
  size_t off = 0;
  auto carve = [&](size_t nbytes) -> void* {
    void* p = (void*)((char*)d_ws + off);
    off += (nbytes + 255) & ~(size_t)255;
    return p;
  };
  int*   degi   = (int*)carve((size_t)N * 4);
  int*   rowptr = (int*)carve((size_t)(N + 1) * 4);
  int*   cursor = (int*)carve((size_t)N * 4);
  float* ew     = (float*)carve((size_t)E * 4);
  v2f*   csr    = (v2f*)carve((size_t)E * 8);
  float* Tx     = (float*)carve((size_t)N * 320 * 4);        // [N, max K*F_in]
  float* hA     = (float*)carve((size_t)N * 152 * 4);
  float* hB     = (float*)carve((size_t)N * 152 * 4);
  float* Wp     = (float*)carve((size_t)3 * GATE_STRIDE * 4); // packed i,c,o
  (void)ws_size; (void)n_in; (void)out_size;

  const int T = 256;
  auto nb = [&](long long n) -> unsigned { return (unsigned)((n + T - 1) / T); };

  // ---- CSR build (once): degree, scan, fill; edge weights
  zero_int_kernel<<<nb(N), T, 0, stream>>>(degi, (unsigned)N);
  zero_int_kernel<<<nb(N), T, 0, stream>>>(cursor, (unsigned)N);
  degree_kernel<<<nb(E), T, 0, stream>>>(edst, degi, E);
  edge_weight_kernel<<<nb(E), T, 0, stream>>>(esrc, edst, degi, ew, E);
  scan_rowptr_kernel<<<1, 1024, 0, stream>>>(degi, rowptr, N);
  csr_fill_kernel<<<nb(E), T, 0, stream>>>(esrc, edst, ew, rowptr, cursor, csr, E);

  const int Karr[4]  = {2, 3, 4, 5};
  const int FinA[4]  = {10, 16, 32, 64};
  const int FoutA[4] = {16, 32, 64, 152};

  const float* hin = x;
  float* hbuf[2] = {hA, hB};

  for (int l = 0; l < 4; ++l) {
    const int K = Karr[l], Fin = FinA[l], Fout = FoutA[l], RD = K * Fin;
    const int Fp = (Fout + 15) & ~15;
    const unsigned totNF = (unsigned)((long long)N * Fin);

    copy_slot0_kernel<<<nb(totNF), T, 0, stream>>>(hin, Tx, totNF, Fin, RD);
    for (int k = 1; k < K; ++k) {
      spmm_gather_kernel<<<nb(totNF), T, 0, stream>>>(
          csr, rowptr, Tx, RD,
          /*koff_src=*/(unsigned)((k - 1) * Fin),
          /*koff_dst=*/(unsigned)(k * Fin),
          /*koff_sub=*/(unsigned)((k >= 2 ? k - 2 : 0) * Fin),
          /*has_sub=*/(k >= 2) ? 1 : 0,
          /*alpha=*/(k == 1) ? -1.0f : -2.0f,
          Fin, totNF);
    }

    // param dict order: per layer, per gate [i,f,c,o]: Wx, bx, bh, b
    const int base = 2 + l * 16;
    const float* Wi  = (const float*)d_in[base + 0 * 4 + 0];
    const float* bxi = (const float*)d_in[base + 0 * 4 + 1];
    const float* bhi = (const float*)d_in[base + 0 * 4 + 2];
    const float* bbi = (const float*)d_in[base + 0 * 4 + 3];
    const float* Wc  = (const float*)d_in[base + 2 * 4 + 0];
    const float* bxc = (const float*)d_in[base + 2 * 4 + 1];
    const float* bhc = (const float*)d_in[base + 2 * 4 + 2];
    const float* bbc = (const float*)d_in[base + 2 * 4 + 3];
    const float* Wo  = (const float*)d_in[base + 3 * 4 + 0];
    const float* bxo = (const float*)d_in[base + 3 * 4 + 1];
    const float* bho = (const float*)d_in[base + 3 * 4 + 2];
    const float* bbo = (const float*)d_in[base + 3 * 4 + 3];
    // f-gate (gi==1) intentionally unused: it multiplies C==0 in the reference.

    const unsigned packTot = (unsigned)(RD * Fp);
    pack_weights_kernel<<<nb(packTot), T, 0, stream>>>(Wi, Wp + 0 * GATE_STRIDE, RD, Fout, Fp);
    pack_weights_kernel<<<nb(packTot), T, 0, stream>>>(Wc, Wp + 1 * GATE_STRIDE, RD, Fout, Fp);
    pack_weights_kernel<<<nb(packTot), T, 0, stream>>>(Wo, Wp + 2 * GATE_STRIDE, RD, Fout, Fp);

    dim3 grid((N + 63) / 64, Fp / 16);
    gate_gemm_wmma<<<grid, 128, 0, stream>>>(Tx, RD, Wp,
                                             bxi, bhi, bbi, bxc, bhc, bbc, bxo, bho, bbo,
                                             hbuf[l & 1], N, Fout, Fp);
    hin = hbuf[l & 1];
  }

  const float* linW = (const float*)d_in[66];
  const float* linb = (const float*)d_in[67];
  linear_softmax_kernel<<<nb(N), T, 0, stream>>>(hin, linW, linb, (float*)d_out, N, 152);
}